// GATLinkPredictorFixed_89000312307816
// MI455X (gfx1250) — compile-verified
//
#include <hip/hip_runtime.h>
#include <hip/hip_bf16.h>
#include <math.h>

typedef __attribute__((ext_vector_type(2))) float v2f;
typedef __attribute__((ext_vector_type(8))) float v8f;

#define LEAKY_SLOPE 0.2f

// ---------- monotonic float<->uint key for atomicMax segment-max ----------
__device__ __forceinline__ unsigned f2key(float v) {
  unsigned b = __float_as_uint(v);
  return (b & 0x80000000u) ? ~b : (b | 0x80000000u);
}
__device__ __forceinline__ float key2f(unsigned k) {
  return __uint_as_float((k & 0x80000000u) ? (k ^ 0x80000000u) : ~k);
}
__device__ __forceinline__ float leaky(float v) {
  return v > 0.0f ? v : v * LEAKY_SLOPE;
}

// ---------------------------------------------------------------------------
// Full-precision FP32 WMMA GEMM: C[M,Ncol] = A[M,K] * B[K,Ncol]
// One wave32 per 16x16 output tile, V_WMMA_F32_16X16X4_F32 over K in steps of 4.
// M, Ncol multiples of 16 (N=100000=6250*16; Ncol in {128,32}); K=128.
// A 16x4 layout: lanes 0-15 row=r hold K={k,k+1}; lanes 16-31 same rows K={k+2,k+3}.
// B 4x16 layout: lanes 0-15 col=r hold K={k,k+1}; lanes 16-31 K={k+2,k+3}.
// C/D: vgpr i -> row (i + 8*half), col r.
// ---------------------------------------------------------------------------
__global__ void wmma_gemm_f32(const float* __restrict__ A,
                              const float* __restrict__ B,
                              float* __restrict__ C,
                              int K, int Ncol, int totalTiles) {
  int wave = blockIdx.x * (blockDim.x >> 5) + (threadIdx.x >> 5);
  if (wave >= totalTiles) return;          // wave-uniform: EXEC stays all-ones
  int tilesPerRow = Ncol >> 4;
  int tm = wave / tilesPerRow;
  int tn = wave % tilesPerRow;
  int lane = threadIdx.x & 31;
  int half = lane >> 4;                    // 0: lanes 0-15, 1: lanes 16-31
  int r    = lane & 15;

  const float* Arow = A + (size_t)(tm * 16 + r) * K;      // row r of the tile
  const float* Bcol = B + (size_t)(tn * 16 + r);          // col r of the tile

  v8f acc = {};
  for (int k = 0; k < K; k += 4) {
    v2f a = *(const v2f*)(Arow + k + 2 * half);           // 8B aligned b64 load
    v2f b;
    b.x = Bcol[(size_t)(k + 2 * half) * Ncol];
    b.y = Bcol[(size_t)(k + 2 * half + 1) * Ncol];
    acc = __builtin_amdgcn_wmma_f32_16x16x4_f32(
        /*neg_a=*/false, a, /*neg_b=*/false, b,
        /*c_mod=*/(short)0, acc, /*reuse_a=*/false, /*reuse_b=*/false);
  }

  float* Cp = C + (size_t)(tm * 16 + 8 * half) * Ncol + tn * 16 + r;
#pragma unroll
  for (int i = 0; i < 8; ++i) Cp[(size_t)i * Ncol] = acc[i];
}

// ---------------------------------------------------------------------------
// att_s[n,h] = dot(h[n,h,:], a_s[h,:]) ; att_d likewise. One thread per (n,h).
// ---------------------------------------------------------------------------
__global__ void att_compute(const float* __restrict__ Hfeat,
                            const float* __restrict__ a_s,
                            const float* __restrict__ a_d,
                            float* __restrict__ att_s,
                            float* __restrict__ att_d,
                            int n, int Hh, int Dd) {
  int t = blockIdx.x * blockDim.x + threadIdx.x;
  if (t >= n * Hh) return;
  int node = t / Hh, h = t % Hh;
  const float* hp = Hfeat + (size_t)node * Hh * Dd + h * Dd;
  const float* asv = a_s + h * Dd;
  const float* adv = a_d + h * Dd;
  float ss = 0.0f, sd = 0.0f;
  for (int c = 0; c < Dd; ++c) {
    float x = hp[c];
    ss += x * asv[c];
    sd += x * adv[c];
  }
  att_s[t] = ss;
  att_d[t] = sd;
}

// ---------------------------------------------------------------------------
// Edge pass 1: segment-max of leaky_relu(att_s[src]+att_d[dst]) via uint keys.
// One thread per (edge, head). Edges e < E0 come from src/dst; e >= E0 are
// self-loops (node e-E0).
// ---------------------------------------------------------------------------
__global__ void edge_max(const int* __restrict__ src, const int* __restrict__ dst,
                         int E0, int ET,
                         const float* __restrict__ att_s,
                         const float* __restrict__ att_d,
                         unsigned* __restrict__ emax_bits, int Hh) {
  int t = blockIdx.x * blockDim.x + threadIdx.x;
  if (t >= ET * Hh) return;
  int e = t / Hh, h = t % Hh;
  int s = (e < E0) ? src[e] : (e - E0);
  int d = (e < E0) ? dst[e] : (e - E0);
  float v = leaky(att_s[s * Hh + h] + att_d[d * Hh + h]);
  atomicMax(&emax_bits[d * Hh + h], f2key(v));
}

// ---------------------------------------------------------------------------
// Edge pass 2: denom[dst,h] += exp(e - emax[dst,h]). One thread per (edge, head).
// ---------------------------------------------------------------------------
__global__ void edge_sum(const int* __restrict__ src, const int* __restrict__ dst,
                         int E0, int ET,
                         const float* __restrict__ att_s,
                         const float* __restrict__ att_d,
                         const unsigned* __restrict__ emax_bits,
                         float* __restrict__ denom, int Hh) {
  int t = blockIdx.x * blockDim.x + threadIdx.x;
  if (t >= ET * Hh) return;
  int e = t / Hh, h = t % Hh;
  int s = (e < E0) ? src[e] : (e - E0);
  int d = (e < E0) ? dst[e] : (e - E0);
  float v = leaky(att_s[s * Hh + h] + att_d[d * Hh + h]);
  float m = key2f(emax_bits[d * Hh + h]);
  atomicAdd(&denom[d * Hh + h], expf(v - m));
}

// ---------------------------------------------------------------------------
// Edge pass 3: out[dst,h,:] += h[src,h,:] * alpha.
// ONE WAVE per (edge, head): lane = channel (Dd == 32 == wave32 width), so the
// h[src] gather is a single contiguous 128B load per wave and the scatter-adds
// are a contiguous 128B atomic burst — this pass carries ~1.7GB and sets the
// kernel's runtime, so coalescing here is what the 23.3 TB/s HBM rewards.
// Per-(edge,head) scalars are same-address across lanes (broadcast loads).
// ---------------------------------------------------------------------------
__global__ void edge_agg(const int* __restrict__ src, const int* __restrict__ dst,
                         int E0, int ET,
                         const float* __restrict__ att_s,
                         const float* __restrict__ att_d,
                         const unsigned* __restrict__ emax_bits,
                         const float* __restrict__ denom,
                         const float* __restrict__ Hfeat,
                         float* __restrict__ Out, int Hh, int Dd) {
  int t = blockIdx.x * blockDim.x + threadIdx.x;
  int c = t & 31;        // channel within head (Dd == 32)
  int p = t >> 5;        // (edge, head) pair
  if (p >= ET * Hh) return;
  int e = p / Hh, h = p % Hh;
  int s = (e < E0) ? src[e] : (e - E0);
  int d = (e < E0) ? dst[e] : (e - E0);
  float v = leaky(att_s[s * Hh + h] + att_d[d * Hh + h]);
  float m = key2f(emax_bits[d * Hh + h]);
  float alpha = expf(v - m) / denom[d * Hh + h];
  float hv = Hfeat[(size_t)s * Hh * Dd + h * Dd + c];
  atomicAdd(&Out[(size_t)d * Hh * Dd + h * Dd + c], hv * alpha);
}

// out = elu(out + b)   (layer-1 epilogue, flattened over H*D channels)
__global__ void bias_elu(float* __restrict__ x, const float* __restrict__ b,
                         int total, int C) {
  int t = blockIdx.x * blockDim.x + threadIdx.x;
  if (t >= total) return;
  float v = x[t] + b[t % C];
  x[t] = v > 0.0f ? v : (expf(v) - 1.0f);
}

// out = out + b   (layer-2 epilogue)
__global__ void bias_add(float* __restrict__ x, const float* __restrict__ b,
                         int total, int C) {
  int t = blockIdx.x * blockDim.x + threadIdx.x;
  if (t >= total) return;
  x[t] = x[t] + b[t % C];
}

extern "C" void kernel_launch(void* const* d_in, const int* in_sizes, int n_in,
                              void* d_out, int out_size, void* d_ws, size_t ws_size,
                              hipStream_t stream) {
  const float* features = (const float*)d_in[0];
  const int*   src      = (const int*)d_in[1];
  const int*   dst      = (const int*)d_in[2];
  const float* W1       = (const float*)d_in[3];
  const float* a1s      = (const float*)d_in[4];
  const float* a1d      = (const float*)d_in[5];
  const float* b1       = (const float*)d_in[6];
  const float* W2       = (const float*)d_in[7];
  const float* a2s      = (const float*)d_in[8];
  const float* a2d      = (const float*)d_in[9];
  const float* b2       = (const float*)d_in[10];

  const int IN = 128, Hh = 4, Dd = 32, HD = 128;
  const int Nn = in_sizes[0] / IN;      // 100000
  const int E0 = in_sizes[1];           // 1600000
  const int ET = E0 + Nn;               // + self-loops

  // ---- workspace carve-up (floats) ----
  float* ws = (float*)d_ws;
  size_t off = 0;
  float*    h1     = ws + off; off += (size_t)Nn * HD;   // [N,128]
  float*    atts1  = ws + off; off += (size_t)Nn * Hh;
  float*    attd1  = ws + off; off += (size_t)Nn * Hh;
  unsigned* emax1  = (unsigned*)(ws + off); off += (size_t)Nn * Hh;
  float*    denom1 = ws + off; off += (size_t)Nn * Hh;
  float*    out1   = ws + off; off += (size_t)Nn * HD;   // layer-1 output / layer-2 input
  float*    h2     = ws + off; off += (size_t)Nn * Dd;   // [N,32]
  float*    atts2  = ws + off; off += (size_t)Nn;
  float*    attd2  = ws + off; off += (size_t)Nn;
  unsigned* emax2  = (unsigned*)(ws + off); off += (size_t)Nn;
  float*    denom2 = ws + off; off += (size_t)Nn;
  float*    out2   = (float*)d_out;                      // [N,32]

  const int TB = 256;
  const int wavesPerBlock = TB / 32;

  // ================= Layer 1 =================
  {
    int tiles = (Nn / 16) * (HD / 16);
    wmma_gemm_f32<<<(tiles + wavesPerBlock - 1) / wavesPerBlock, TB, 0, stream>>>(
        features, W1, h1, IN, HD, tiles);
  }
  att_compute<<<(Nn * Hh + TB - 1) / TB, TB, 0, stream>>>(h1, a1s, a1d, atts1, attd1, Nn, Hh, Dd);

  hipMemsetAsync(emax1, 0, (size_t)Nn * Hh * sizeof(unsigned), stream);  // key 0 < any float
  hipMemsetAsync(denom1, 0, (size_t)Nn * Hh * sizeof(float), stream);
  hipMemsetAsync(out1, 0, (size_t)Nn * HD * sizeof(float), stream);

  {
    long long nt = (long long)ET * Hh;
    edge_max<<<(unsigned)((nt + TB - 1) / TB), TB, 0, stream>>>(src, dst, E0, ET, atts1, attd1, emax1, Hh);
    edge_sum<<<(unsigned)((nt + TB - 1) / TB), TB, 0, stream>>>(src, dst, E0, ET, atts1, attd1, emax1, denom1, Hh);
    long long na = nt * 32;          // one lane per (edge, head, channel)
    edge_agg<<<(unsigned)((na + TB - 1) / TB), TB, 0, stream>>>(src, dst, E0, ET, atts1, attd1,
                                                                emax1, denom1, h1, out1, Hh, Dd);
  }
  bias_elu<<<(Nn * HD + TB - 1) / TB, TB, 0, stream>>>(out1, b1, Nn * HD, HD);

  // ================= Layer 2 =================
  {
    int tiles = (Nn / 16) * (Dd / 16);
    wmma_gemm_f32<<<(tiles + wavesPerBlock - 1) / wavesPerBlock, TB, 0, stream>>>(
        out1, W2, h2, HD, Dd, tiles);
  }
  att_compute<<<(Nn + TB - 1) / TB, TB, 0, stream>>>(h2, a2s, a2d, atts2, attd2, Nn, 1, Dd);

  hipMemsetAsync(emax2, 0, (size_t)Nn * sizeof(unsigned), stream);
  hipMemsetAsync(denom2, 0, (size_t)Nn * sizeof(float), stream);
  hipMemsetAsync(out2, 0, (size_t)Nn * Dd * sizeof(float), stream);

  {
    long long nt = (long long)ET;
    edge_max<<<(unsigned)((nt + TB - 1) / TB), TB, 0, stream>>>(src, dst, E0, ET, atts2, attd2, emax2, 1);
    edge_sum<<<(unsigned)((nt + TB - 1) / TB), TB, 0, stream>>>(src, dst, E0, ET, atts2, attd2, emax2, denom2, 1);
    long long na = nt * 32;
    edge_agg<<<(unsigned)((na + TB - 1) / TB), TB, 0, stream>>>(src, dst, E0, ET, atts2, attd2,
                                                                emax2, denom2, h2, out2, 1, Dd);
  }
  bias_add<<<(Nn * Dd + TB - 1) / TB, TB, 0, stream>>>(out2, b2, Nn * Dd, Dd);
}